// BatchedSemiAttention_48318382080001
// MI455X (gfx1250) — compile-verified
//
#include <hip/hip_runtime.h>
#include <hip/hip_bf16.h>
#include <math.h>

// Problem constants (match reference)
#define L_DIM   4
#define G_NODES 10000
#define E_EDGES 160000
#define INP_D   128
#define KEY_D   64
#define VAL_D   128
#define OUTC    256        // [K(64) | Q(64) | V(128)] concatenated per (node, layer)
#define LN_EPS  1e-5f

typedef __attribute__((ext_vector_type(2))) float v2f;
typedef __attribute__((ext_vector_type(8))) float v8f;

// ---------------------------------------------------------------------------
// Kernel 0: pack Wk|Wq|Wv into the WMMA B-operand layout.
// Wpack[tile(16)][kk(32)][lane(32)] = float2{ W[k0][c], W[k0+1][c] }
//   where k0 = kk*4 + (lane>=16 ? 2 : 0), c = tile*16 + (lane&15).
// In proj, lane's B operand for step kk is one b64 load at base + kk*256 B.
// ---------------------------------------------------------------------------
__global__ __launch_bounds__(256) void pack_kernel(
    const float* __restrict__ Wk, const float* __restrict__ Wq,
    const float* __restrict__ Wv, float2* __restrict__ Wpack)
{
    int idx = blockIdx.x * blockDim.x + threadIdx.x;   // 16*32*32 = 16384
    if (idx >= 16 * 32 * 32) return;
    int lane = idx & 31;
    int kk   = (idx >> 5) & 31;
    int tile = idx >> 10;
    int k0 = kk * 4 + ((lane >> 4) << 1);
    int c  = tile * 16 + (lane & 15);

    const float* W; int stride; int cw;
    if (c < 64)       { W = Wk; stride = KEY_D; cw = c; }
    else if (c < 128) { W = Wq; stride = KEY_D; cw = c - 64; }
    else              { W = Wv; stride = VAL_D; cw = c - 128; }

    float2 p;
    p.x = W[(size_t)k0 * stride + cw];
    p.y = W[(size_t)(k0 + 1) * stride + cw];
    Wpack[idx] = p;
}

// ---------------------------------------------------------------------------
// Kernel 1: fused K/Q/V projection via V_WMMA_F32_16X16X4_F32.
// Grid: 2500 blocks (625 node-tiles x 4 layers), 128 threads (4 waves).
// Each wave computes four 16x16 output tiles (K-dim 128 = 32 WMMA steps).
// Inner loop per step: ds_load_b64 (A) + global_load_b64 imm-offset (B) + wmma.
// Output layout: kqv[g][l][0:256] node-major for locality of edge gathers.
// ---------------------------------------------------------------------------
#define XS_STRIDE 132   // padded LDS row stride (floats) to avoid bank conflicts

__global__ __launch_bounds__(128) void proj_kernel(
    const float* __restrict__ x,
    const float* __restrict__ bk, const float* __restrict__ bq,
    const float* __restrict__ bv, const float2* __restrict__ Wpack,
    float* __restrict__ kqv)
{
    __shared__ float xs[16 * XS_STRIDE];   // 16 rows x 128 cols, padded

    const int block = blockIdx.x;
    const int l     = block & 3;           // layer
    const int g0    = (block >> 2) * 16;   // first node of tile

    // --- stage 16x128 x-tile into LDS (coalesced float4) ---
    const float4* xsrc = reinterpret_cast<const float4*>(
        x + ((size_t)l * G_NODES + g0) * INP_D);
    for (int i = threadIdx.x; i < 512; i += 128) {
        int row  = i >> 5;            // 32 float4 per row
        int col4 = i & 31;
        *reinterpret_cast<float4*>(&xs[row * XS_STRIDE + col4 * 4]) = xsrc[i];
    }
    __syncthreads();

    const int wave    = threadIdx.x >> 5;
    const int lane    = threadIdx.x & 31;
    const int halfsel = lane >> 4;        // 0: K 0..1, 1: K 2..3 (A/B layout)
    const int lane16  = lane & 15;
    const float* xrow = &xs[lane16 * XS_STRIDE + halfsel * 2]; // A lane base

    for (int t = 0; t < 4; ++t) {
        const int nt = wave * 4 + t;      // N-tile 0..15 (wave-uniform)
        const int c0 = nt * 16;           // output column base 0..255

        // bias: C[m][n] depends only on n = c0 + lane16 (wave-uniform branch)
        const float* bvec; int cw0;
        if (c0 < 64)       { bvec = bk; cw0 = c0; }
        else if (c0 < 128) { bvec = bq; cw0 = c0 - 64; }
        else               { bvec = bv; cw0 = c0 - 128; }
        const float bias = bvec[cw0 + lane16];

        // per-lane packed-B base: Wpack[nt][0][lane]; step kk -> +kk*256 bytes
        const float2* Bt = Wpack + ((size_t)nt << 10) + lane;

        v8f acc;
        #pragma unroll
        for (int r = 0; r < 8; ++r) acc[r] = bias;

        #pragma unroll
        for (int kk = 0; kk < 32; ++kk) {
            v2f a;
            a[0] = xrow[kk * 4];          // ds_load_b64, imm offset kk*16
            a[1] = xrow[kk * 4 + 1];
            float2 bp = Bt[kk * 32];      // global_load_b64, imm offset kk*256
            v2f bmat; bmat[0] = bp.x; bmat[1] = bp.y;
            acc = __builtin_amdgcn_wmma_f32_16x16x4_f32(
                false, a, false, bmat, (short)0, acc, false, false);
        }

        // store: VGPR r holds row m = r + halfsel*8, col c0 + lane16
        #pragma unroll
        for (int r = 0; r < 8; ++r) {
            const int g = g0 + r + halfsel * 8;
            kqv[((size_t)g * L_DIM + l) * OUTC + c0 + lane16] = acc[r];
        }
    }
}

// ---------------------------------------------------------------------------
// Kernel 2: CSR offsets from sorted row_indices (lower_bound per node).
// ---------------------------------------------------------------------------
__global__ void offsets_kernel(const int* __restrict__ rows,
                               int* __restrict__ offsets)
{
    int g = blockIdx.x * blockDim.x + threadIdx.x;
    if (g > G_NODES) return;
    if (g == G_NODES) { offsets[G_NODES] = E_EDGES; return; }
    int lo = 0, hi = E_EDGES;
    while (lo < hi) {
        int mid = (lo + hi) >> 1;
        if (rows[mid] < g) lo = mid + 1; else hi = mid;
    }
    offsets[g] = lo;
}

// ---------------------------------------------------------------------------
// Kernel 3: per-(edge, layer) attention logits w = <k_col, q_col>.
// wbuf layout [E][4] so the aggregator reads one float4 per edge.
// ---------------------------------------------------------------------------
__global__ __launch_bounds__(256) void score_kernel(
    const float* __restrict__ kqv, const int* __restrict__ cols,
    float* __restrict__ wbuf)
{
    int idx = blockIdx.x * blockDim.x + threadIdx.x;
    if (idx >= E_EDGES * L_DIM) return;
    int e = idx >> 2;
    int l = idx & 3;
    int col = cols[e];
    const float4* kp = reinterpret_cast<const float4*>(
        kqv + ((size_t)col * L_DIM + l) * OUTC);      // k at [0:16), q at [16:32) f4
    float dot = 0.f;
    #pragma unroll
    for (int i = 0; i < 16; ++i) {
        float4 kv = kp[i];
        float4 qv = kp[16 + i];
        dot += kv.x * qv.x + kv.y * qv.y + kv.z * qv.z + kv.w * qv.w;
    }
    wbuf[e * 4 + l] = dot;
}

// ---------------------------------------------------------------------------
// Kernel 4: one wave32 per destination node: segment softmax + weighted
// V aggregation + SiLU + LayerNorm, fused write of [L,G,VAL] output.
// ---------------------------------------------------------------------------
__global__ __launch_bounds__(128) void agg_kernel(
    const float* __restrict__ kqv, const int* __restrict__ cols,
    const int* __restrict__ offsets, const float* __restrict__ wbuf,
    const float* __restrict__ gamma, const float* __restrict__ beta,
    float* __restrict__ out)
{
    const int wave = threadIdx.x >> 5;
    const int lane = threadIdx.x & 31;
    const int g = blockIdx.x * 4 + wave;
    if (g >= G_NODES) return;

    const int start = offsets[g];
    const int end   = offsets[g + 1];
    const float4* w4 = reinterpret_cast<const float4*>(wbuf);
    const float4* v4 = reinterpret_cast<const float4*>(kqv);

    // --- pass 1: per-layer max over incident edges (lane-parallel) ---
    float mx0 = -3.4e38f, mx1 = -3.4e38f, mx2 = -3.4e38f, mx3 = -3.4e38f;
    for (int e = start + lane; e < end; e += 32) {
        float4 w = w4[e];
        mx0 = fmaxf(mx0, w.x); mx1 = fmaxf(mx1, w.y);
        mx2 = fmaxf(mx2, w.z); mx3 = fmaxf(mx3, w.w);
    }
    #pragma unroll
    for (int off = 16; off >= 1; off >>= 1) {
        mx0 = fmaxf(mx0, __shfl_xor(mx0, off, 32));
        mx1 = fmaxf(mx1, __shfl_xor(mx1, off, 32));
        mx2 = fmaxf(mx2, __shfl_xor(mx2, off, 32));
        mx3 = fmaxf(mx3, __shfl_xor(mx3, off, 32));
    }

    // --- pass 2: exp-sum + weighted V accumulation; lanes stripe VAL=128 ---
    float4 acc[4];
    #pragma unroll
    for (int l = 0; l < 4; ++l) acc[l] = make_float4(0.f, 0.f, 0.f, 0.f);
    float ps0 = 0.f, ps1 = 0.f, ps2 = 0.f, ps3 = 0.f;

    for (int e = start; e < end; ++e) {
        const int col = cols[e];
        const float4 w = w4[e];                       // broadcast load
        const float p0 = __expf(w.x - mx0);
        const float p1 = __expf(w.y - mx1);
        const float p2 = __expf(w.z - mx2);
        const float p3 = __expf(w.w - mx3);
        ps0 += p0; ps1 += p1; ps2 += p2; ps3 += p3;
        const size_t vb = (size_t)col * (L_DIM * 64); // float4 units per node
        float pl[4] = {p0, p1, p2, p3};
        #pragma unroll
        for (int l = 0; l < 4; ++l) {
            float4 v = v4[vb + (size_t)l * 64 + 32 + lane];  // V at f4 ofs 32
            acc[l].x += pl[l] * v.x; acc[l].y += pl[l] * v.y;
            acc[l].z += pl[l] * v.z; acc[l].w += pl[l] * v.w;
        }
    }

    const float4 gm = reinterpret_cast<const float4*>(gamma)[lane];
    const float4 bt = reinterpret_cast<const float4*>(beta)[lane];
    const float psum[4] = {ps0, ps1, ps2, ps3};
    const bool has_edges = (end > start);

    #pragma unroll
    for (int l = 0; l < 4; ++l) {
        const float inv = has_edges ? 1.f / psum[l] : 0.f;
        float4 s = acc[l];
        s.x *= inv; s.y *= inv; s.z *= inv; s.w *= inv;
        // SiLU: x * sigmoid(x) = x / (1 + e^-x)
        s.x = s.x / (1.f + __expf(-s.x));
        s.y = s.y / (1.f + __expf(-s.y));
        s.z = s.z / (1.f + __expf(-s.z));
        s.w = s.w / (1.f + __expf(-s.w));
        // LayerNorm over VAL=128 (wave reduce)
        float lsum = s.x + s.y + s.z + s.w;
        #pragma unroll
        for (int off = 16; off >= 1; off >>= 1)
            lsum += __shfl_xor(lsum, off, 32);
        const float mu = lsum * (1.f / 128.f);
        float4 d;
        d.x = s.x - mu; d.y = s.y - mu; d.z = s.z - mu; d.w = s.w - mu;
        float lsq = d.x * d.x + d.y * d.y + d.z * d.z + d.w * d.w;
        #pragma unroll
        for (int off = 16; off >= 1; off >>= 1)
            lsq += __shfl_xor(lsq, off, 32);
        const float rstd = rsqrtf(lsq * (1.f / 128.f) + LN_EPS);
        float4 y;
        y.x = d.x * rstd * gm.x + bt.x;
        y.y = d.y * rstd * gm.y + bt.y;
        y.z = d.z * rstd * gm.z + bt.z;
        y.w = d.w * rstd * gm.w + bt.w;
        reinterpret_cast<float4*>(out)[((size_t)l * G_NODES + g) * 32 + lane] = y;
    }
}

// ---------------------------------------------------------------------------
// Launch: pack (B layout) -> proj (WMMA GEMM) -> offsets (CSR) ->
//         score (logits) -> agg (fused softmax/agg/SiLU/LN).
// Workspace: Wpack 131,072 B | kqv 40,960,000 B | wbuf 2,560,000 B |
//            offsets 40,004 B  (total ~43.7 MB)
// ---------------------------------------------------------------------------
extern "C" void kernel_launch(void* const* d_in, const int* in_sizes, int n_in,
                              void* d_out, int out_size, void* d_ws, size_t ws_size,
                              hipStream_t stream) {
    (void)in_sizes; (void)n_in; (void)out_size; (void)ws_size;
    const float* x     = (const float*)d_in[0];
    const float* Wk    = (const float*)d_in[1];
    const float* bk    = (const float*)d_in[2];
    const float* Wq    = (const float*)d_in[3];
    const float* bq    = (const float*)d_in[4];
    const float* Wv    = (const float*)d_in[5];
    const float* bv    = (const float*)d_in[6];
    const float* gamma = (const float*)d_in[7];
    const float* beta  = (const float*)d_in[8];
    const int*   rows  = (const int*)d_in[9];
    const int*   colsi = (const int*)d_in[10];
    float* out = (float*)d_out;

    char* ws = (char*)d_ws;
    float2* Wpack  = (float2*)ws;                       // 131,072 B
    float*  kqv    = (float*)(ws + 131072);             // [G][L][256] f32
    float*  wbuf   = (float*)(ws + 131072 + 40960000);  // [E][4]      f32
    int*    offsets= (int*)  (ws + 131072 + 43520000);  // [G+1]       i32

    pack_kernel<<<64, 256, 0, stream>>>(Wk, Wq, Wv, Wpack);
    proj_kernel<<<(G_NODES / 16) * L_DIM, 128, 0, stream>>>(
        x, bk, bq, bv, Wpack, kqv);
    offsets_kernel<<<(G_NODES + 1 + 255) / 256, 256, 0, stream>>>(rows, offsets);
    score_kernel<<<(E_EDGES * L_DIM + 255) / 256, 256, 0, stream>>>(
        kqv, colsi, wbuf);
    agg_kernel<<<(G_NODES + 3) / 4, 128, 0, stream>>>(
        kqv, colsi, offsets, wbuf, gamma, beta, out);
}